// SGCEvaluator_86792699117682
// MI455X (gfx1250) — compile-verified
//
#include <hip/hip_runtime.h>
#include <hip/hip_bf16.h>

typedef __attribute__((ext_vector_type(2))) float v2f;
typedef __attribute__((ext_vector_type(8))) float v8f;

#define F_DIM 128

// ---------- degree / norm kernels ----------
__global__ void init_deg_kernel(float* __restrict__ deg, int N) {
    int i = blockIdx.x * blockDim.x + threadIdx.x;
    if (i < N) deg[i] = 1.0f;  // self-loop
}

__global__ void count_deg_kernel(const int* __restrict__ dst, float* __restrict__ deg, int E) {
    int e = blockIdx.x * blockDim.x + threadIdx.x;
    if (e < E) unsafeAtomicAdd(&deg[dst[e]], 1.0f);
}

__global__ void rsqrt_kernel(float* __restrict__ deg, int N) {
    int i = blockIdx.x * blockDim.x + threadIdx.x;
    if (i < N) deg[i] = rsqrtf(deg[i]);
}

// ---------- hop: self-loop seed (h_out = dinv^2 * h_in) ----------
__global__ void selfloop_kernel(const float* __restrict__ hin, const float* __restrict__ dinv,
                                float* __restrict__ hout, int N) {
    int i = blockIdx.x * blockDim.x + threadIdx.x;  // over N*32 float4 chunks
    if (i >= N * 32) return;
    int n = i >> 5;
    int c = i & 31;
    float s = dinv[n];
    s = s * s;
    float4 v = ((const float4*)hin)[(size_t)n * 32 + c];
    float4 r;
    r.x = s * v.x; r.y = s * v.y; r.z = s * v.z; r.w = s * v.w;
    ((float4*)hout)[(size_t)n * 32 + c] = r;
}

// ---------- hop: edge scatter (wave per edge, lane = 4 features) ----------
__global__ void edge_scatter_kernel(const int* __restrict__ src, const int* __restrict__ dst,
                                    const float* __restrict__ dinv,
                                    const float* __restrict__ hin, float* __restrict__ hout,
                                    int E) {
    int wave = (blockIdx.x * blockDim.x + threadIdx.x) >> 5;
    int lane = threadIdx.x & 31;
    if (wave >= E) return;
    int s = src[wave];
    int d = dst[wave];
    float coef = dinv[s] * dinv[d];
    float4 v = ((const float4*)(hin + (size_t)s * F_DIM))[lane];
    float* ho = hout + (size_t)d * F_DIM + lane * 4;
    unsafeAtomicAdd(ho + 0, coef * v.x);
    unsafeAtomicAdd(ho + 1, coef * v.y);
    unsafeAtomicAdd(ho + 2, coef * v.z);
    unsafeAtomicAdd(ho + 3, coef * v.w);
}

// ---------- final GEMM: out[N x C] = h[N x 128] @ W[C x 128]^T + b, f32 WMMA ----------
// One wave computes one 16(M) x 16(N) tile with 32x v_wmma_f32_16x16x4_f32 over K=128.
// Out-of-range B columns (wcol >= C) clamp to a valid W row: their accumulator
// columns are garbage but never stored, and this keeps the K loop branch-free
// (pure b64 loads + wmma, full EXEC throughout).
__global__ void gemm_wmma_kernel(const float* __restrict__ h, const float* __restrict__ W,
                                 const float* __restrict__ bias, float* __restrict__ out,
                                 int N, int C) {
    int wave = (blockIdx.x * blockDim.x + threadIdx.x) >> 5;
    int lane = threadIdx.x & 31;
    int tilesM = (N + 15) >> 4;
    int tilesN = (C + 15) >> 4;
    int tm = wave / tilesN;           // wave-uniform
    int tn = (wave - tm * tilesN) * 16;
    if (tm >= tilesM) return;         // wave-uniform exit: EXEC stays all-1s for WMMA
    int row = tm * 16;

    int half = lane >> 4;             // K-half selector for A/B
    int lid  = lane & 15;             // M index (A) / N index (B,C,D)

    int arow = row + lid; if (arow > N - 1) arow = N - 1;   // clamp (N % 16 == 0 normally)
    int wcol = tn + lid;
    bool wvalid = (wcol < C);
    int bcol = wvalid ? wcol : (C - 1);                     // clamp: finite data, dead column

    const v2f* aptr = (const v2f*)(h + (size_t)arow * F_DIM);  // v2f-indexed (8B aligned)
    const v2f* bptr = (const v2f*)(W + (size_t)bcol * F_DIM);

    v8f acc = {};
#pragma unroll
    for (int k0 = 0; k0 < F_DIM; k0 += 4) {
        int idx = (k0 >> 1) + half;   // fragment: floats {k0+2*half, k0+2*half+1}
        v2f a = aptr[idx];
        v2f b = bptr[idx];
        acc = __builtin_amdgcn_wmma_f32_16x16x4_f32(
            /*neg_a=*/false, a, /*neg_b=*/false, b,
            /*c_mod=*/(short)0, acc, /*reuse_a=*/false, /*reuse_b=*/false);
    }

    if (wvalid) {
        float bv = bias[wcol];
#pragma unroll
        for (int r = 0; r < 8; ++r) {
            int m = row + r + 8 * half;
            if (m < N) out[(size_t)m * C + wcol] = acc[r] + bv;
        }
    }
}

extern "C" void kernel_launch(void* const* d_in, const int* in_sizes, int n_in,
                              void* d_out, int out_size, void* d_ws, size_t ws_size,
                              hipStream_t stream) {
    const float* x  = (const float*)d_in[0];
    const int*   ei = (const int*)d_in[1];   // [src(E), dst(E)]
    const float* W  = (const float*)d_in[2];
    const float* b  = (const float*)d_in[3];
    float* out = (float*)d_out;

    const int F = F_DIM;
    const int N = in_sizes[0] / F;
    const int E = in_sizes[1] / 2;
    const int C = in_sizes[3];

    const int* src = ei;
    const int* dst = ei + E;

    // workspace: dinv (N floats, 256B aligned) | h1 (N*F) | h2 (N*F)
    char* ws = (char*)d_ws;
    float* dinv = (float*)ws;
    size_t off = ((size_t)N * sizeof(float) + 255) & ~(size_t)255;
    float* h1 = (float*)(ws + off);
    float* h2 = h1 + (size_t)N * F;

    const int B = 256;

    // 1) degrees (incl. self-loop) -> dinv = rsqrt(deg)
    init_deg_kernel<<<(N + B - 1) / B, B, 0, stream>>>(dinv, N);
    count_deg_kernel<<<(E + B - 1) / B, B, 0, stream>>>(dst, dinv, E);
    rsqrt_kernel<<<(N + B - 1) / B, B, 0, stream>>>(dinv, N);

    // 2) hop 1: x -> h1
    selfloop_kernel<<<((size_t)N * 32 + B - 1) / B, B, 0, stream>>>(x, dinv, h1, N);
    edge_scatter_kernel<<<((size_t)E * 32 + B - 1) / B, B, 0, stream>>>(src, dst, dinv, x, h1, E);

    // 3) hop 2: h1 -> h2
    selfloop_kernel<<<((size_t)N * 32 + B - 1) / B, B, 0, stream>>>(h1, dinv, h2, N);
    edge_scatter_kernel<<<((size_t)E * 32 + B - 1) / B, B, 0, stream>>>(src, dst, dinv, h1, h2, E);

    // 4) WMMA GEMM: h2 @ W^T + b -> out
    int tilesM = (N + 15) / 16;
    int tilesN = (C + 15) / 16;
    size_t waves = (size_t)tilesM * tilesN;
    size_t threads = waves * 32;
    gemm_wmma_kernel<<<(threads + B - 1) / B, B, 0, stream>>>(h2, W, b, out, N, C);
}